// SmithWatermanLoss_755914244505
// MI455X (gfx1250) — compile-verified
//
#include <hip/hip_runtime.h>
#include <hip/hip_bf16.h>

// ---------------------------------------------------------------------------
// Smith-Waterman differentiable loss, fused for MI455X (gfx1250, wave32).
//   Kernel 1: score matrix x[b,i,j] via V_WMMA_F32_16X16X4_F32 (K=4 matmul)
//   Kernel 2: anti-diagonal DP (one WGP per batch) + fused streaming LSE
//   Kernel 3: -mean over batches -> d_out[0]
// ---------------------------------------------------------------------------

typedef float v2f __attribute__((ext_vector_type(2)));
typedef float v8f __attribute__((ext_vector_type(8)));

#define BATCH    32
#define LEN      512
#define AC       511              // A = C = LEN-1
#define NDIAG    1021             // n = A + C - 1
#define MSLOT    511              // m = (A+C)/2
#define GAP_OPEN (-4.0f)
#define GAP_EXT  (-1.0f)
#define NEGINF   (-1e30f)

// ---------------- Kernel 1: x = pred @ onehot(tgt)^T via WMMA --------------
// One wave (32 lanes) per 16x16 output tile; 8 waves per block.
// A layout (f32 16x4): lane l holds A[l%16, 2*(l/16)+v], v=0..1
// B layout (f32 4x16): lane l holds B[2*(l/16)+v, l%16]
// D layout (f32 16x16): VGPR v -> row (v + 8*(l/16)), col l%16
__global__ __launch_bounds__(256)
void sw_score_wmma(const float* __restrict__ pred,
                   const int*   __restrict__ tgt,
                   float*       __restrict__ x) {
    const int wave = threadIdx.x >> 5;
    const int lane = threadIdx.x & 31;
    const int tile = blockIdx.x * 8 + wave;        // b*1024 + ti*32 + tj
    const int tj   = tile & 31;
    const int ti   = (tile >> 5) & 31;
    const int b    = tile >> 10;

    const int half = lane >> 4;                    // 0 or 1
    const int mn   = lane & 15;

    // A fragment: predictions[b, ti*16+mn, 2*half + {0,1}]
    const float* prow = pred + (((b * LEN) + ti * 16 + mn) << 2) + 2 * half;
    v2f a;
    a.x = prow[0];
    a.y = prow[1];

    // B fragment: onehot(tgt[b, tj*16+mn]) at channels 2*half + {0,1}
    const int t = tgt[b * LEN + tj * 16 + mn];
    v2f bm;
    bm.x = (t == 2 * half)     ? 1.0f : 0.0f;
    bm.y = (t == 2 * half + 1) ? 1.0f : 0.0f;

    v8f c = {};
    // D = A x B + 0   (8 args: neg_a, A, neg_b, B, c_mod, C, reuse_a, reuse_b)
    c = __builtin_amdgcn_wmma_f32_16x16x4_f32(false, a, false, bm,
                                              (short)0, c, false, false);

    float* xout = x + ((long)b << 18)
                    + (long)(ti * 16 + 8 * half) * LEN
                    + tj * 16 + mn;
#pragma unroll
    for (int v = 0; v < 8; ++v)
        xout[(long)v * LEN] = c[v];
}

// ---------------- log-sum-exp helpers (finite NEG_INF arithmetic) ----------
__device__ __forceinline__ float lse2(float a, float b) {
    float M = fmaxf(a, b);
    return M + __logf(__expf(a - M) + __expf(b - M));
}
__device__ __forceinline__ float lse3(float a, float b, float c) {
    float M = fmaxf(fmaxf(a, b), c);
    return M + __logf(__expf(a - M) + __expf(b - M) + __expf(c - M));
}
__device__ __forceinline__ float lse4(float a, float b, float c, float d) {
    float M = fmaxf(fmaxf(a, b), fmaxf(c, d));
    return M + __logf(__expf(a - M) + __expf(b - M) +
                      __expf(c - M) + __expf(d - M));
}
__device__ __forceinline__ void acc_lse(float& rm, float& rs, float v) {
    if (v > rm) { rs = rs * __expf(rm - v) + 1.0f; rm = v; }
    else        { rs += __expf(v - rm); }
}

// ---------------- Kernel 2: wavefront DP + fused final reduction -----------
// One block per batch; thread j owns wavefront slot j. h2/h1/h0 rotate
// through 3 LDS phases; one workgroup barrier per anti-diagonal.
__global__ __launch_bounds__(512)
void sw_dp(const float* __restrict__ x, float* __restrict__ finals) {
    const int b = blockIdx.x;
    const int j = threadIdx.x;

    __shared__ float h[3][512][3];
    __shared__ float red_m[512];
    __shared__ float red_s[512];

#pragma unroll
    for (int p = 0; p < 3; ++p) {
        h[p][j][0] = NEGINF; h[p][j][1] = NEGINF; h[p][j][2] = NEGINF;
    }
    __syncthreads();

    const float* xb = x + ((long)b << 18);
    const bool active = (j < MSLOT);

    float rm = NEGINF;   // streaming LSE running max
    float rs = 0.0f;     // sum of exp(v - rm)

    for (int d = 0; d < NDIAG; ++d) {
        const int p2 = d % 3;          // h2 (two diagonals back)
        const int p1 = (d + 1) % 3;    // h1 (previous diagonal)
        const int p0 = (d + 2) % 3;    // h0 (output)

        // slot j on diagonal d <-> cell (r, c) of the 511x511 score grid
        const int  r     = ((510 + d) >> 1) - j;
        const int  c     = d - r;
        const bool valid = active && (r >= 0) && (r < AC) && (c >= 0) && (c < AC);
        const float s    = valid ? xb[r * LEN + c] : NEGINF;

        float h0_0 = NEGINF, h0_1 = NEGINF, h0_2 = NEGINF;
        if (active) {
            const float h2_0 = h[p2][j][0], h2_1 = h[p2][j][1], h2_2 = h[p2][j][2];
            const float h1_0 = h[p1][j][0], h1_1 = h[p1][j][1], h1_2 = h[p1][j][2];

            float r0, r1, d0, d1, d2;
            if ((d & 1) == 0) {        // smo == 1
                r0 = (j > 0) ? h[p1][j - 1][0] : NEGINF;
                r1 = (j > 0) ? h[p1][j - 1][1] : NEGINF;
                d0 = h1_0; d1 = h1_1; d2 = h1_2;
            } else {                   // smo == 0
                r0 = h1_0; r1 = h1_1;
                d0 = (j < MSLOT - 1) ? h[p1][j + 1][0] : NEGINF;
                d1 = (j < MSLOT - 1) ? h[p1][j + 1][1] : NEGINF;
                d2 = (j < MSLOT - 1) ? h[p1][j + 1][2] : NEGINF;
            }

            h0_0 = lse4(h2_0 + s, h2_1 + s, h2_2 + s, s);
            h0_1 = lse2(r0 + GAP_OPEN, r1 + GAP_EXT);
            h0_2 = lse3(d0 + GAP_OPEN, d1 + GAP_OPEN, d2 + GAP_EXT);

            h[p0][j][0] = h0_0; h[p0][j][1] = h0_1; h[p0][j][2] = h0_2;
        }

        // fused final reduction: hij[b,r,c,*] + x[b, r+1, c+1]
        if (valid) {
            const float xp = xb[(r + 1) * LEN + (c + 1)];
            acc_lse(rm, rs, h0_0 + xp);
            acc_lse(rm, rs, h0_1 + xp);
            acc_lse(rm, rs, h0_2 + xp);
        }
        __syncthreads();   // publish p0 before next step reads it as h1
    }

    // block-wide combine of (rm, rs) streaming-LSE states
    red_m[j] = rm; red_s[j] = rs;
    __syncthreads();
    for (int off = 256; off >= 1; off >>= 1) {
        if (j < off) {
            const float m1 = red_m[j],       s1 = red_s[j];
            const float m2 = red_m[j + off], s2 = red_s[j + off];
            const float M  = fmaxf(m1, m2);
            red_m[j] = M;
            red_s[j] = s1 * __expf(m1 - M) + s2 * __expf(m2 - M);
        }
        __syncthreads();
    }
    if (j == 0)
        finals[b] = red_m[0] + __logf(red_s[0]);   // logsumexp for batch b
}

// ---------------- Kernel 3: loss = -mean_b(final[b]) -----------------------
__global__ void sw_finalize(const float* __restrict__ finals,
                            float* __restrict__ out) {
    if (threadIdx.x == 0 && blockIdx.x == 0) {
        float s = 0.0f;
#pragma unroll
        for (int b = 0; b < BATCH; ++b) s += finals[b];
        out[0] = -s * (1.0f / (float)BATCH);
    }
}

// ---------------------------------------------------------------------------
extern "C" void kernel_launch(void* const* d_in, const int* in_sizes, int n_in,
                              void* d_out, int out_size, void* d_ws, size_t ws_size,
                              hipStream_t stream) {
    const float* pred = (const float*)d_in[0];   // (32, 512, 4) f32
    const int*   tgt  = (const int*)d_in[1];     // (32, 512)    i32
    float*       out  = (float*)d_out;           // scalar loss

    float* xbuf   = (float*)d_ws;                          // 32*512*512 f32 (33.5 MB)
    float* finals = xbuf + (size_t)BATCH * LEN * LEN;      // 32 f32

    // 32*32*32 tiles of 16x16, 8 waves (tiles) per 256-thread block
    sw_score_wmma<<<(BATCH * 32 * 32) / 8, 256, 0, stream>>>(pred, tgt, xbuf);

    // one workgroup per batch, one thread per wavefront slot
    sw_dp<<<BATCH, 512, 0, stream>>>(xbuf, finals);

    sw_finalize<<<1, 32, 0, stream>>>(finals, out);
}